// CorrelationProj_62560493634035
// MI455X (gfx1250) — compile-verified
//
#include <hip/hip_runtime.h>
#include <hip/hip_bf16.h>

// CorrelationProj for MI455X (gfx1250), wave32.
// One wave per (b, l, query-pixel): 25 nearest-neighbor offsets on the WMMA M
// dimension, channel reduction (C=128) via 32 x V_WMMA_F32_16X16X4_F32 steps.
// Branch-free inner loop: gathers always use clipped (in-bounds) addresses,
// validity applied at store time via a wave-ballot mask (reference semantics:
// corr = dot(gather_clipped, q) * valid). Query vector is staged in LDS
// pre-scaled by 1/C and broadcast across all 16 N columns of B so every D
// column carries the same dot product (robust to B's N striping).

typedef float v2f __attribute__((ext_vector_type(2)));
typedef float v8f __attribute__((ext_vector_type(8)));

#define Bn 2
#define Ln 4
#define Cn 128
#define Hn 48
#define Wn 48
#define Kn 25   // (2*MD/STRIDE + 1)^2 = 5x5
#define HW (Hn * Wn)

__global__ __launch_bounds__(256) void CorrelationProj_gfx1250_kernel(
    const float* __restrict__ input1,    // [B,C,H,W]
    const float* __restrict__ input2,    // [B,L,C,H2,W2]
    const float* __restrict__ qcoords,   // [B,H,W,3]
    const float* __restrict__ scoords,   // [B,L,H2,W2,3]
    const float* __restrict__ sceneP,    // [B,L,3,4]
    float* __restrict__ corr_out,        // [B, L*K, H, W]
    float* __restrict__ coords_out)      // [B, L*K, H, W, 3]
{
    __shared__ float qs[8][Cn];          // per-wave staged query vector (pre-scaled)

    const int lane = threadIdx.x & 31;
    const int wib  = threadIdx.x >> 5;
    const int wid  = blockIdx.x * 8 + wib;        // [0, B*L*H*W)
    const int qpix = wid % HW;
    const int bl   = wid / HW;
    const int l    = bl % Ln;
    const int b    = bl / Ln;
    const int h    = qpix / Wn;
    const int w    = qpix % Wn;

    // ---- stage query vector into LDS, folded 1/C scale: dot(g, q/C) = corr ----
    const float* q_base = input1 + (size_t)b * Cn * HW + h * Wn + w;
    #pragma unroll
    for (int j = 0; j < 4; ++j) {
        const int c = lane + 32 * j;
        qs[wib][c] = q_base[(size_t)c * HW] * (1.0f / Cn);
    }
    __syncthreads();

    // ---- project query 3D point into scene frame l (wave-uniform) ----
    const float* qc = qcoords + (((size_t)b * Hn + h) * Wn + w) * 3;
    const float x = qc[0], y = qc[1], z = qc[2];
    const float* P = sceneP + ((size_t)b * Ln + l) * 12;
    const float p0 = P[0] * x + P[1] * y + P[2]  * z + P[3];
    const float p1 = P[4] * x + P[5] * y + P[6]  * z + P[7];
    const float p2 = P[8] * x + P[9] * y + P[10] * z + P[11] + 1e-8f;
    const float uc = floorf(p0 / p2 + 0.5f);
    const float vc = floorf(p1 / p2 + 0.5f);

    const float* g_base = input2 + ((size_t)b * Ln + l) * Cn * HW;
    const bool hi    = lane >= 16;
    const int  khalf = hi ? 2 : 0;       // A layout: VGPR0 = K{0|2}, VGPR1 = K{1|3}

    float* corr_bl = corr_out + ((size_t)b * Ln + l) * Kn * HW + h * Wn + w;

    #pragma unroll
    for (int tile = 0; tile < 2; ++tile) {
        const int m  = lane & 15;                     // A-matrix row in this lane
        const int k  = tile * 16 + m;                 // neighborhood offset index
        const int kc = (k < Kn) ? k : (Kn - 1);       // clamp padded rows (never stored)
        const float dx = (float)(kc % 5) - 2.0f;
        const float dy = (float)(kc / 5) - 2.0f;
        const float sx = uc + dx, sy = vc + dy;
        const bool rvalid = (sx >= 0.f) && (sx <= (float)(Wn - 1)) &&
                            (sy >= 0.f) && (sy <= (float)(Hn - 1));
        // wave-uniform validity mask: bit m <-> row tile*16+m (low half of wave)
#if __has_builtin(__builtin_amdgcn_ballot_w32)
        const unsigned vmask = __builtin_amdgcn_ballot_w32(rvalid);
#else
        const unsigned vmask = (unsigned)__ballot(rvalid);
#endif
        const int xi = (int)fminf(fmaxf(sx, 0.f), (float)(Wn - 1));
        const int yi = (int)fminf(fmaxf(sy, 0.f), (float)(Hn - 1));
        const float* gptr = g_base + (yi * Wn + xi);  // always in-bounds

        v8f acc = {0.f, 0.f, 0.f, 0.f, 0.f, 0.f, 0.f, 0.f};
        #pragma unroll 8
        for (int c0 = 0; c0 < Cn; c0 += 4) {
            const int ca = c0 + khalf;
            v2f a;
            a.x = gptr[(size_t)ca       * HW];        // unconditional gathers
            a.y = gptr[(size_t)(ca + 1) * HW];
            const v2f bb = *(const v2f*)&qs[wib][ca]; // 8B-aligned broadcast B
            acc = __builtin_amdgcn_wmma_f32_16x16x4_f32(
                /*neg_a=*/false, a, /*neg_b=*/false, bb,
                /*c_mod=*/(short)0, acc, /*reuse_a=*/false, /*reuse_b=*/false);
        }

        // D layout: lane n<16 holds rows 0..7 (col n); lane 16+n holds rows 8..15.
        // Every column is identical (B broadcast) -> read column 0 halves.
        if (lane == 0 || lane == 16) {
            const int rbase = hi ? 8 : 0;
            #pragma unroll
            for (int r = 0; r < 8; ++r) {
                const int kk = tile * 16 + rbase + r;
                if (kk < Kn) {
                    const float v = (vmask & (1u << (rbase + r))) ? acc[r] : 0.0f;
                    corr_bl[(size_t)kk * HW] = v;
                }
            }
        }
    }

    // ---- gathered scene coordinates: lanes 0..24, one k each ----
    if (lane < Kn) {
        const int k = lane;
        const float dx = (float)(k % 5) - 2.0f;
        const float dy = (float)(k / 5) - 2.0f;
        const float sx = uc + dx, sy = vc + dy;
        const bool valid = (sx >= 0.f) && (sx <= (float)(Wn - 1)) &&
                           (sy >= 0.f) && (sy <= (float)(Hn - 1));
        const int xi = (int)fminf(fmaxf(sx, 0.f), (float)(Wn - 1));
        const int yi = (int)fminf(fmaxf(sy, 0.f), (float)(Hn - 1));
        const float* sc = scoords +
            (((size_t)b * Ln + l) * HW + yi * Wn + xi) * 3;
        const float cx = valid ? sc[0] : 0.f;
        const float cy = valid ? sc[1] : 0.f;
        const float cz = valid ? sc[2] : 0.f;
        float* co = coords_out +
            ((((size_t)b * Ln + l) * Kn + k) * HW + h * Wn + w) * 3;
        co[0] = cx; co[1] = cy; co[2] = cz;
    }
}

extern "C" void kernel_launch(void* const* d_in, const int* in_sizes, int n_in,
                              void* d_out, int out_size, void* d_ws, size_t ws_size,
                              hipStream_t stream) {
    const float* input1  = (const float*)d_in[0];   // [B,C,H,W]
    const float* input2  = (const float*)d_in[1];   // [B,L,C,H2,W2]
    const float* qcoords = (const float*)d_in[2];   // [B,H,W,3]
    const float* scoords = (const float*)d_in[3];   // [B,L,H2,W2,3]
    const float* sceneP  = (const float*)d_in[4];   // [B,L,3,4]

    float* corr   = (float*)d_out;                              // first output
    float* coords = corr + (size_t)Bn * Ln * Kn * HW;           // second output

    const int total_waves = Bn * Ln * HW;           // 18432 waves, one per (b,l,pixel)
    const int blocks = total_waves / 8;             // 256 threads = 8 wave32 per block
    CorrelationProj_gfx1250_kernel<<<blocks, 256, 0, stream>>>(
        input1, input2, qcoords, scoords, sceneP, corr, coords);
}